// LiMEPromptTuning_17325898072236
// MI455X (gfx1250) — compile-verified
//
#include <hip/hip_runtime.h>
#include <hip/hip_bf16.h>

#define BB 32
#define SS 256
#define TT 256
#define HH 4096
#define EE 64
#define KK 4
#define SC_CHUNKS 4
#define S_PER_CHUNK (SS / SC_CHUNKS)
#define EPS_F 1e-6f

typedef __attribute__((ext_vector_type(2))) float v2f;
typedef __attribute__((ext_vector_type(8))) float v8f;

// ---------------------------------------------------------------------------
// Kernel 1a: partial reduction over S (bandwidth-dominant: 128 MB read)
// grid = (HH/1024, BB, SC_CHUNKS), block = 256, float4 per thread
// ---------------------------------------------------------------------------
__global__ void reduce_s_partial(const float* __restrict__ in, float* __restrict__ Xp) {
    const int tid = threadIdx.x;
    const int h   = (blockIdx.x * 256 + tid) * 4;
    const int b   = blockIdx.y;
    const int sc  = blockIdx.z;
    const float* p = in + ((size_t)(b * SS + sc * S_PER_CHUNK)) * HH + h;
    float4 acc = make_float4(0.f, 0.f, 0.f, 0.f);
    for (int s = 0; s < S_PER_CHUNK; ++s) {
        const float4 v = *(const float4*)(p + (size_t)s * HH);
        acc.x += v.x; acc.y += v.y; acc.z += v.z; acc.w += v.w;
    }
    *(float4*)(Xp + ((size_t)(sc * BB + b)) * HH + h) = acc;
}

// ---------------------------------------------------------------------------
// Kernel 1b: fold SC_CHUNKS partials -> X[B][H]
// grid = (HH/1024, BB), block = 256
// ---------------------------------------------------------------------------
__global__ void reduce_partials(const float* __restrict__ Xp, float* __restrict__ X) {
    const int tid = threadIdx.x;
    const int h   = (blockIdx.x * 256 + tid) * 4;
    const int b   = blockIdx.y;
    float4 acc = make_float4(0.f, 0.f, 0.f, 0.f);
    for (int sc = 0; sc < SC_CHUNKS; ++sc) {
        const float4 v = *(const float4*)(Xp + ((size_t)(sc * BB + b)) * HH + h);
        acc.x += v.x; acc.y += v.y; acc.z += v.z; acc.w += v.w;
    }
    *(float4*)(X + (size_t)b * HH + h) = acc;
}

// ---------------------------------------------------------------------------
// Kernel 2: cond[:, :64] = X @ W_proj[:64,:]^T / S  via V_WMMA_F32_16X16X4_F32
// Single block of 256 threads = 8 waves; wave w handles tile (mt = w/4, nt = w%4)
// A layout (16x4 f32): lane = M (both halves), VGPR v holds K = 2*(lane>>4) + v
// B layout (4x16 f32): lane&15 = N,            VGPR v holds K = 2*(lane>>4) + v
// C layout (16x16 f32): VGPR v -> M = v + 8*(lane>>4), N = lane&15
// ---------------------------------------------------------------------------
__global__ void cond_wmma(const float* __restrict__ X, const float* __restrict__ Wp,
                          float* __restrict__ cond) {
    const int tid  = threadIdx.x;
    const int lane = tid & 31;
    const int w    = tid >> 5;   // 0..7
    const int mt   = w >> 2;     // 0..1  (batch tile)
    const int nt   = w & 3;      // 0..3  (expert tile)
    const int half = lane >> 4;  // 0/1
    const int lid  = lane & 15;

    const float* arow = X  + (size_t)(mt * 16 + lid) * HH + 2 * half;
    const float* brow = Wp + (size_t)(nt * 16 + lid) * HH + 2 * half;

    v8f c = {};
#pragma unroll 8
    for (int k = 0; k < HH; k += 4) {
        const v2f a = *(const v2f*)(arow + k);
        const v2f bm = *(const v2f*)(brow + k);
        c = __builtin_amdgcn_wmma_f32_16x16x4_f32(
                /*neg_a=*/false, a, /*neg_b=*/false, bm,
                /*c_mod=*/(short)0, c, /*reuse_a=*/false, /*reuse_b=*/false);
    }

    const float inv = 1.0f / (float)SS;
#pragma unroll
    for (int v = 0; v < 8; ++v) {
        const int brow_g = mt * 16 + v + 8 * half;
        const int ecol   = nt * 16 + lid;
        cond[brow_g * EE + ecol] = c[v] * inv;
    }
}

// ---------------------------------------------------------------------------
// Kernel 3: global abs-max scales. Non-negative float bits compare as uints,
// so uint atomicMax is exact and order-independent (deterministic).
// grid = TT, block = 256
// ---------------------------------------------------------------------------
__global__ void scales_kernel(const float* __restrict__ pb, const float* __restrict__ pd,
                              const float* __restrict__ cond, unsigned int* __restrict__ sc) {
    __shared__ float sh[256];
    __shared__ float sd[256];
    const int t = blockIdx.x;
    const int tid = threadIdx.x;

    float hmax = 0.f;
    for (int i = tid; i < BB * EE; i += 256) {
        const int b = i >> 6, e = i & 63;
        hmax = fmaxf(hmax, fabsf(pb[(size_t)t * HH + e] + cond[b * EE + e]));
    }
    const float dmax = (tid < EE) ? fabsf(pd[(size_t)t * HH + tid]) : 0.f;

    sh[tid] = hmax; sd[tid] = dmax;
    __syncthreads();
    for (int off = 128; off > 0; off >>= 1) {
        if (tid < off) {
            sh[tid] = fmaxf(sh[tid], sh[tid + off]);
            sd[tid] = fmaxf(sd[tid], sd[tid + off]);
        }
        __syncthreads();
    }
    if (tid == 0) {
        atomicMax(sc + 0, __float_as_uint(sh[0]));
        atomicMax(sc + 1, __float_as_uint(sd[0]));
    }
}

// ---------------------------------------------------------------------------
// Kernel 4: routing (top-4 of 64, softmax-denominator cancels under renorm)
// + streamed output (128 MB write). grid = (TT, BB), block = 256
// ---------------------------------------------------------------------------
__global__ void route_out(const float* __restrict__ pb, const float* __restrict__ pd,
                          const float* __restrict__ limes, const float* __restrict__ lshared,
                          const float* __restrict__ gamma, const float* __restrict__ cond,
                          const unsigned int* __restrict__ sc, float* __restrict__ out) {
    __shared__ float sl[EE];
    __shared__ float swv[KK];
    __shared__ int   sidx[KK];
    __shared__ float sg;

    const int t = blockIdx.x;
    const int b = blockIdx.y;
    const int tid = threadIdx.x;

    const float Hs = fmaxf(__uint_as_float(sc[0]), EPS_F);
    const float Ds = fmaxf(__uint_as_float(sc[1]), EPS_F);

    if (tid < EE) {
        const float hv = pb[(size_t)t * HH + tid] + cond[b * EE + tid];
        sl[tid] = 0.5f * (hv / Hs) + 0.5f * (pd[(size_t)t * HH + tid] / Ds);
    }
    __syncthreads();

    if (tid == 0) {
        float vals[KK]; int idxs[KK]; float lmax0 = 0.f;
        for (int k = 0; k < KK; ++k) {
            float m = -__builtin_inff(); int mi = 0;
            for (int e = 0; e < EE; ++e) {
                if (sl[e] > m) { m = sl[e]; mi = e; }   // ties -> lowest index (matches top_k)
            }
            vals[k] = m; idxs[k] = mi; sl[mi] = -__builtin_inff();
            if (k == 0) lmax0 = m;
        }
        float sum = 0.f;
        for (int k = 0; k < KK; ++k) { vals[k] = expf(vals[k] - lmax0); sum += vals[k]; }
        const float inv = 1.0f / fmaxf(sum, 1e-9f);
        for (int k = 0; k < KK; ++k) { swv[k] = vals[k] * inv; sidx[k] = idxs[k]; }
        sg = 1.0f / (1.0f + expf(-gamma[0]));
    }
    __syncthreads();

    const float g = sg, og = 1.0f - sg;
    const float w0 = swv[0], w1 = swv[1], w2 = swv[2], w3 = swv[3];
    const float* L0 = limes + (size_t)sidx[0] * HH;
    const float* L1 = limes + (size_t)sidx[1] * HH;
    const float* L2 = limes + (size_t)sidx[2] * HH;
    const float* L3 = limes + (size_t)sidx[3] * HH;
    const float* pbr = pb + (size_t)t * HH;
    const float* pdr = pd + (size_t)t * HH;
    float* outr = out + ((size_t)(b * TT + t)) * HH;

#pragma unroll
    for (int i = 0; i < HH / (256 * 4); ++i) {
        const int h = (i * 256 + tid) * 4;
        const float4 pbv = *(const float4*)(pbr + h);
        const float4 pdv = *(const float4*)(pdr + h);
        const float4 l0  = *(const float4*)(L0 + h);
        const float4 l1  = *(const float4*)(L1 + h);
        const float4 l2  = *(const float4*)(L2 + h);
        const float4 l3  = *(const float4*)(L3 + h);
        const float4 shv = *(const float4*)(lshared + h);
        float4 o;
        o.x = (pbv.x + pdv.x) * (og * (w0*l0.x + w1*l1.x + w2*l2.x + w3*l3.x) + g * shv.x);
        o.y = (pbv.y + pdv.y) * (og * (w0*l0.y + w1*l1.y + w2*l2.y + w3*l3.y) + g * shv.y);
        o.z = (pbv.z + pdv.z) * (og * (w0*l0.z + w1*l1.z + w2*l2.z + w3*l3.z) + g * shv.z);
        o.w = (pbv.w + pdv.w) * (og * (w0*l0.w + w1*l1.w + w2*l2.w + w3*l3.w) + g * shv.w);
        *(float4*)(outr + h) = o;
    }
}

// ---------------------------------------------------------------------------
// Host-side launch. Workspace layout (bytes):
//   [0)                 Xp   : SC_CHUNKS*BB*HH floats  (2 MB)
//   [2 MB)              X    : BB*HH floats            (512 KB)
//   [2.5 MB)            cond : BB*EE floats            (8 KB)
//   [2.5 MB + 8 KB)     scales: 2 x uint               (8 B)
// ---------------------------------------------------------------------------
extern "C" void kernel_launch(void* const* d_in, const int* in_sizes, int n_in,
                              void* d_out, int out_size, void* d_ws, size_t ws_size,
                              hipStream_t stream) {
    const float* input_embeds = (const float*)d_in[0];
    const float* prompt_base  = (const float*)d_in[1];
    const float* prompt_delta = (const float*)d_in[2];
    const float* limes        = (const float*)d_in[3];
    const float* lime_shared  = (const float*)d_in[4];
    const float* gamma        = (const float*)d_in[5];
    const float* w_proj       = (const float*)d_in[6];
    float* out = (float*)d_out;

    char* ws = (char*)d_ws;
    float* Xp   = (float*)(ws);
    float* X    = (float*)(ws + (size_t)SC_CHUNKS * BB * HH * sizeof(float));
    float* cond = (float*)(ws + (size_t)(SC_CHUNKS + 1) * BB * HH * sizeof(float));
    unsigned int* scales =
        (unsigned int*)(ws + (size_t)(SC_CHUNKS + 1) * BB * HH * sizeof(float)
                           + (size_t)BB * EE * sizeof(float));

    // 1a. partial S-reduction
    reduce_s_partial<<<dim3(HH / 1024, BB, SC_CHUNKS), 256, 0, stream>>>(input_embeds, Xp);
    // 1b. fold partials
    reduce_partials<<<dim3(HH / 1024, BB), 256, 0, stream>>>(Xp, X);
    // 2. cond[:, :E] via f32 WMMA (single workgroup, 8 waves = 2x4 tiles)
    cond_wmma<<<1, 256, 0, stream>>>(X, w_proj, cond);
    // 3. abs-max scales (init to 0 first; uint max on non-negative float bits)
    hipMemsetAsync(scales, 0, 2 * sizeof(unsigned int), stream);
    scales_kernel<<<TT, 256, 0, stream>>>(prompt_base, prompt_delta, cond, scales);
    // 4. routing + output stream
    route_out<<<dim3(TT, BB), 256, 0, stream>>>(prompt_base, prompt_delta, limes,
                                                lime_shared, gamma, cond, scales, out);
}